// GCNDecoder_18614388261507
// MI455X (gfx1250) — compile-verified
//
#include <hip/hip_runtime.h>
#include <hip/hip_bf16.h>

typedef __attribute__((ext_vector_type(2))) float v2f;
typedef __attribute__((ext_vector_type(8))) float v8f;

// ---------------- degree / normalization ----------------

__global__ void init_deg_kernel(float* __restrict__ deg, int n) {
  int i = blockIdx.x * blockDim.x + threadIdx.x;
  if (i < n) deg[i] = 1.0f;  // self-loop contributes 1 to every dst degree
}

__global__ void count_deg_kernel(const int* __restrict__ dst, float* __restrict__ deg, int E) {
  int e = blockIdx.x * blockDim.x + threadIdx.x;
  if (e < E)
    __hip_atomic_fetch_add(&deg[dst[e]], 1.0f, __ATOMIC_RELAXED, __HIP_MEMORY_SCOPE_AGENT);
}

__global__ void rsqrt_kernel(float* __restrict__ deg, int n) {
  int i = blockIdx.x * blockDim.x + threadIdx.x;
  if (i < n) deg[i] = rsqrtf(deg[i]);  // deg >= 1 always (self-loop)
}

// ---------------- fp32 WMMA GEMM: out[N,NC] = A[N,K] @ W[K,NC] ----------------
// One wave -> one 16x16 output tile. K-loop in steps of 4 using
// V_WMMA_F32_16X16X4_F32. Requires N % 16 == 0 (50000 = 3125*16).
//
// Fragment layouts (ISA 7.12.2):
//  A 16x4 : element (m,k): lane = (k>=2)*16 + m, vgpr = k&1  -> contiguous float2/lane
//  B 4x16 : vgpr r: lanes 0-15 hold K=r, lanes 16-31 hold K=2+r, N = lane%16
//  C/D    : vgpr r: lanes 0-15 hold M=r, lanes 16-31 hold M=8+r, N = lane%16

template <int K, int NC>
__global__ void gemm_wmma_kernel(const float* __restrict__ A,
                                 const float* __restrict__ W,
                                 float* __restrict__ out) {
  const int lane = threadIdx.x & 31;
  const int wave = threadIdx.x >> 5;
  const int m = lane & 15;         // row within tile (A) / col within tile (B,C)
  const int h = lane >> 4;         // lane-half selects K pair / M half
  const int rowTile = blockIdx.x * 16;
  const int colTile = (blockIdx.y * 4 + wave) * 16;

  const float* arow = A + (size_t)(rowTile + m) * K + 2 * h;
  const float* wcol = W + (size_t)(2 * h) * NC + colTile + m;

  v8f c = {0.f, 0.f, 0.f, 0.f, 0.f, 0.f, 0.f, 0.f};

#pragma unroll
  for (int k0 = 0; k0 < K; k0 += 4) {
    float2 av = *(const float2*)(arow + k0);
    v2f a = {av.x, av.y};
    const float* wp = wcol + (size_t)k0 * NC;
    v2f b = {wp[0], wp[NC]};
    // 8 args: (neg_a, A, neg_b, B, c_mod, C, reuse_a, reuse_b)
    c = __builtin_amdgcn_wmma_f32_16x16x4_f32(false, a, false, b, (short)0, c,
                                              false, false);
  }

  float* orow = out + (size_t)(rowTile + h * 8) * NC + colTile + m;
#pragma unroll
  for (int r = 0; r < 8; ++r) orow[(size_t)r * NC] = c[r];
}

// ---------------- aggregation ----------------

// out[v,f] = dinv[v]^2 * xw[v,f] + bias[f]   (self-loop term + bias)
__global__ void self_init_kernel(const float* __restrict__ xw,
                                 const float* __restrict__ dinv,
                                 const float* __restrict__ bias,
                                 float* __restrict__ out, int total, int F) {
  int i = blockIdx.x * blockDim.x + threadIdx.x;
  if (i >= total) return;
  int v = i / F;
  int f = i - v * F;
  float s = dinv[v];
  out[i] = s * s * xw[i] + bias[f];
}

// One wave per edge: out[dst] += dinv[src]*dinv[dst] * xw[src]
__global__ void edge_agg_kernel(const int* __restrict__ src,
                                const int* __restrict__ dst,
                                const float* __restrict__ dinv,
                                const float* __restrict__ xw,
                                float* __restrict__ out, int E, int F) {
  int wid = (blockIdx.x * blockDim.x + threadIdx.x) >> 5;
  int lane = threadIdx.x & 31;
  if (wid >= E) return;
  int s = src[wid];
  int d = dst[wid];
  float norm = dinv[s] * dinv[d];
  const float* xr = xw + (size_t)s * F;
  float* orow = out + (size_t)d * F;
  for (int f = lane; f < F; f += 32) {
    __hip_atomic_fetch_add(&orow[f], norm * xr[f], __ATOMIC_RELAXED,
                           __HIP_MEMORY_SCOPE_AGENT);
  }
}

__global__ void tanh_kernel(float* __restrict__ out, int n) {
  int i = blockIdx.x * blockDim.x + threadIdx.x;
  if (i < n) out[i] = tanhf(out[i]);
}

// ---------------- launch ----------------

extern "C" void kernel_launch(void* const* d_in, const int* in_sizes, int n_in,
                              void* d_out, int out_size, void* d_ws, size_t ws_size,
                              hipStream_t stream) {
  const float* x  = (const float*)d_in[0];
  const int*   ei = (const int*)d_in[1];
  const float* W1 = (const float*)d_in[2];
  const float* b1 = (const float*)d_in[3];
  const float* W2 = (const float*)d_in[4];
  const float* b2 = (const float*)d_in[5];

  const int DIN  = in_sizes[3];          // 128 (len b1)
  const int DOUT = in_sizes[5];          // 64  (len b2)
  const int N    = in_sizes[0] / DIN;    // 50000
  const int E    = in_sizes[1] / 2;      // 640000
  const int* src = ei;
  const int* dst = ei + E;

  char* ws = (char*)d_ws;
  size_t off = ((size_t)N * sizeof(float) + 255) & ~(size_t)255;
  float* dinv = (float*)ws;
  float* xw1  = (float*)(ws + off);                             // [N, DIN]
  size_t off2 = off + (((size_t)N * DIN * sizeof(float) + 255) & ~(size_t)255);
  float* h1   = (float*)(ws + off2);                            // [N, DIN]
  float* xw2  = xw1;                                            // reuse slot: [N, DOUT]
  float* out  = (float*)d_out;                                  // [N, DOUT]

  const int TB = 256;

  // degrees -> dinv (in place)
  init_deg_kernel<<<(N + TB - 1) / TB, TB, 0, stream>>>(dinv, N);
  count_deg_kernel<<<(E + TB - 1) / TB, TB, 0, stream>>>(dst, dinv, E);
  rsqrt_kernel<<<(N + TB - 1) / TB, TB, 0, stream>>>(dinv, N);

  // ---- layer 1: h1 = Anorm @ (x @ W1) + b1 ----
  gemm_wmma_kernel<128, 128><<<dim3(N / 16, 128 / 64), 128, 0, stream>>>(x, W1, xw1);
  self_init_kernel<<<(N * DIN + TB - 1) / TB, TB, 0, stream>>>(xw1, dinv, b1, h1,
                                                               N * DIN, DIN);
  edge_agg_kernel<<<((size_t)E * 32 + TB - 1) / TB, TB, 0, stream>>>(src, dst, dinv,
                                                                     xw1, h1, E, DIN);

  // ---- layer 2: out = tanh(Anorm @ (h1 @ W2) + b2) ----
  gemm_wmma_kernel<128, 64><<<dim3(N / 16, 64 / 64), 128, 0, stream>>>(h1, W2, xw2);
  self_init_kernel<<<(N * DOUT + TB - 1) / TB, TB, 0, stream>>>(xw2, dinv, b2, out,
                                                                N * DOUT, DOUT);
  edge_agg_kernel<<<((size_t)E * 32 + TB - 1) / TB, TB, 0, stream>>>(src, dst, dinv,
                                                                     xw2, out, E, DOUT);
  tanh_kernel<<<(out_size + TB - 1) / TB, TB, 0, stream>>>(out, out_size);
}